// NeighborhoodAttention_80066780332534
// MI455X (gfx1250) — compile-verified
//
#include <hip/hip_runtime.h>
#include <math.h>

#define GSZ  256
#define PADW 2
#define GP   (GSZ + 2*PADW)   // 260 padded grid
#define EMB  128
#define HID  256
#define NB   2                // batch

typedef __attribute__((ext_vector_type(2))) float v2f;
typedef __attribute__((ext_vector_type(8))) float v8f;

#if __has_builtin(__builtin_amdgcn_wmma_f32_16x16x4_f32)
#define HAVE_WMMA_F32X4 1
#else
#define HAVE_WMMA_F32X4 0
#endif

__device__ __forceinline__ v8f wmma_acc_f32(v2f a, v2f b, v8f c) {
#if HAVE_WMMA_F32X4
  // 8-arg pattern: (neg_a, A, neg_b, B, c_mod, C, reuse_a, reuse_b)
  return __builtin_amdgcn_wmma_f32_16x16x4_f32(false, a, false, b, (short)0, c, false, false);
#else
  c[0] += a[0] * b[0];  // placeholder so a missing builtin is visible as wmma==0
  return c;
#endif
}

// One 16x16 f32 tile: A (16 rows, lda) from LDS, B (KxN row-major, col base pre-applied).
__device__ __forceinline__ v8f wmma_tile_f32(const float* A, int lda,
                                             const float* B, int ldb,
                                             int K, int lane) {
  const int half = (lane >> 4) & 1;
  const int l    = lane & 15;
  v8f acc = {};
  for (int k0 = 0; k0 < K; k0 += 4) {
    const int ka = k0 + half * 2;
    v2f a, b;
    a[0] = A[l * lda + ka];
    a[1] = A[l * lda + ka + 1];
    b[0] = B[(size_t)ka * ldb + l];
    b[1] = B[(size_t)(ka + 1) * ldb + l];
    acc = wmma_acc_f32(a, b, acc);
  }
  return acc;
}

__device__ __forceinline__ void store_tile_lds(float* D, int ldd, v8f acc,
                                               const float* bias, int n0,
                                               int lane, bool relu) {
  const int half = lane >> 4, l = lane & 15;
  const float bv = bias[n0 + l];
#pragma unroll
  for (int vv = 0; vv < 8; ++vv) {
    float x = acc[vv] + bv;
    if (relu) x = fmaxf(x, 0.0f);
    D[(vv + half * 8) * ldd + n0 + l] = x;
  }
}

// interleaved sin/cos 2d embedding, y-first (cols 0..63 from y, 64..127 from x)
__device__ __forceinline__ float posemb_elem(float px, float py, int col) {
  const float p = (col < 64) ? py : px;
  const int jj = col & 63;
  const int k = jj >> 1;
  const float t = powf(10000.0f, (float)k * (1.0f / 32.0f));
  const float val = p * 6.28318530717958647692f / t;
  return (jj & 1) ? cosf(val) : sinf(val);
}

// ---------------- value-index map (idx+1; 0 = missing -> zero row) ---------
__global__ void __launch_bounds__(256)
map_kernel(const int* __restrict__ ctr_coor, int* __restrict__ vmap, int V) {
  const int v = blockIdx.x * 256 + threadIdx.x;
  if (v >= V) return;
  const int b  = ctr_coor[3 * v + 0];
  const int ix = (ctr_coor[3 * v + 1] >> 1) + 128 + PADW;  // //STRIDE - OFFSET + PAD
  const int iy = (ctr_coor[3 * v + 2] >> 1) + 128 + PADW;
  if (b >= 0 && b < NB && ix >= 0 && ix < GP && iy >= 0 && iy < GP)
    vmap[((size_t)b * GP + ix) * GP + iy] = v + 1;
}

// ---------------- qpos: posemb -> 128x256 relu -> 256x128 ------------------
__global__ void __launch_bounds__(64)
qpos_kernel(const float* __restrict__ ref_pts,
            const float* __restrict__ w1, const float* __restrict__ b1,
            const float* __restrict__ w2, const float* __restrict__ b2,
            float* __restrict__ qpos) {
  __shared__ float sEmb[32 * EMB];
  __shared__ float sHid[32 * HID];
  const int tid  = threadIdx.x;
  const int lane = tid & 31;
  const int wv   = tid >> 5;
  const int rowBase = blockIdx.x * 32;

  // pull weights toward L2/WGP$ while we compute the embeddings
  __builtin_prefetch(w1 + (size_t)tid * 512, 0, 1);
  __builtin_prefetch(w2 + (size_t)tid * 512, 0, 1);

  {  // thread t fills row t/2, column half t&1 (each wave fills its own rows)
    const int r = tid >> 1;
    const int row = rowBase + r;
    const float x = ref_pts[row * 3 + 1];
    const float y = ref_pts[row * 3 + 2];
    const float px = (x + 51.2f) / 102.4f;
    const float py = (y + 51.2f) / 102.4f;
    const int c0 = (tid & 1) * 64;
    for (int j = 0; j < 64; ++j)
      sEmb[r * EMB + c0 + j] = posemb_elem(px, py, c0 + j);
  }
  __syncthreads();

  const float* Aemb = sEmb + wv * 16 * EMB;
  float* Hrow = sHid + wv * 16 * HID;
  for (int nt = 0; nt < HID / 16; ++nt) {
    v8f acc = wmma_tile_f32(Aemb, EMB, w1 + nt * 16, HID, EMB, lane);
    store_tile_lds(Hrow, HID, acc, b1, nt * 16, lane, true);
  }
  __syncthreads();

  float* Dst = qpos + (size_t)(rowBase + wv * 16) * EMB;
  for (int nt = 0; nt < EMB / 16; ++nt) {
    v8f acc = wmma_tile_f32(Hrow, HID, w2 + nt * 16, EMB, HID, lane);
    const int half = lane >> 4, l = lane & 15;
    const float bv = b2[nt * 16 + l];
#pragma unroll
    for (int vv = 0; vv < 8; ++vv)
      Dst[(size_t)(vv + half * 8) * EMB + nt * 16 + l] = acc[vv] + bv;
  }
}

// ------- vpos: posemb -> MLP, SE gate from ctr_feat, gated store -----------
__global__ void __launch_bounds__(64)
vpos_kernel(const int* __restrict__ ctr_coor, const float* __restrict__ ctr_feat,
            const float* __restrict__ w1, const float* __restrict__ b1,
            const float* __restrict__ w2, const float* __restrict__ b2,
            const float* __restrict__ se_wr, const float* __restrict__ se_br,
            const float* __restrict__ se_we, const float* __restrict__ se_be,
            float* __restrict__ vposg) {
  __shared__ float sEmb[32 * EMB];   // emb, then reused to hold vpos
  __shared__ float sHid[32 * HID];   // hidden; halves reused as feat / SE-hidden
  const int tid  = threadIdx.x;
  const int lane = tid & 31;
  const int wv   = tid >> 5;
  const int rowBase = blockIdx.x * 32;

  __builtin_prefetch(w1 + (size_t)tid * 512, 0, 1);
  __builtin_prefetch(w2 + (size_t)tid * 512, 0, 1);
  __builtin_prefetch(se_wr + (size_t)tid * 256, 0, 1);
  __builtin_prefetch(se_we + (size_t)tid * 256, 0, 1);

  // the SE-input feature tile is loaded by an async global->LDS copy after the MLP;
  // it lands in the first half of sHid (sFeat), over the dead hidden data.
  // (generic pointers to __shared__ carry the LDS byte offset in their
  //  low 32 bits, which is exactly what the async VDST operand wants)
  float* sFeat = sHid;               // 32*128 floats (overlaps wave0 hidden)
  float* sSeH  = sHid + 32 * EMB;    // 32*128 floats (overlaps wave1 hidden)
  // NOTE: sFeat overlaps the MLP hidden buffer, so the async copy must go to
  // sEmb? No: it must not overwrite live data. Stage it into its own region
  // is impossible within 48KB, so issue the async copy *after* the MLP below.

  {
    const int r = tid >> 1;
    const int row = rowBase + r;
    const int cx = ctr_coor[row * 3 + 1];
    const int cy = ctr_coor[row * 3 + 2];
    const float px = (((float)cx + 0.5f) * 0.2f + 51.2f) / 102.4f;
    const float py = (((float)cy + 0.5f) * 0.2f + 51.2f) / 102.4f;
    const int c0 = (tid & 1) * 64;
    for (int j = 0; j < 64; ++j)
      sEmb[r * EMB + c0 + j] = posemb_elem(px, py, c0 + j);
  }
  __syncthreads();

  const float* Aemb = sEmb + wv * 16 * EMB;
  float* Hrow = sHid + wv * 16 * HID;
  for (int nt = 0; nt < HID / 16; ++nt) {
    v8f acc = wmma_tile_f32(Aemb, EMB, w1 + nt * 16, HID, EMB, lane);
    store_tile_lds(Hrow, HID, acc, b1, nt * 16, lane, true);
  }
  __syncthreads();

  // layer2 -> vpos into sEmb (wave touches only its own 16-row slice)
  float* Vrow = sEmb + wv * 16 * EMB;
  for (int nt = 0; nt < EMB / 16; ++nt) {
    v8f acc = wmma_tile_f32(Hrow, HID, w2 + nt * 16, EMB, HID, lane);
    store_tile_lds(Vrow, EMB, acc, b2, nt * 16, lane, false);
  }
  __syncthreads();

  // SE gate input: async global->LDS copy of the 32x128 ctr_feat tile
  // (GLOBAL_LOAD_ASYNC_TO_LDS_B128, tracked on ASYNCcnt; no VGPR data path)
  {
#pragma unroll
    for (int c = 0; c < 16; ++c) {
      const int idx = c * 64 + tid;  // float4 index within the 32x128 tile
      const float* g = ctr_feat + (size_t)rowBase * EMB + (size_t)idx * 4;
      const unsigned int l = (unsigned int)(size_t)sFeat + (unsigned int)idx * 16u;
      asm volatile("global_load_async_to_lds_b128 %0, %1, off"
                   :: "v"(l), "v"(g) : "memory");
    }
#if __has_builtin(__builtin_amdgcn_s_wait_asynccnt)
    __builtin_amdgcn_s_wait_asynccnt(0);
#else
    asm volatile("s_wait_asynccnt 0x0" ::: "memory");
#endif
  }
  __syncthreads();

  const float* Afeat = sFeat + wv * 16 * EMB;
  float* SeRow = sSeH + wv * 16 * EMB;
  for (int nt = 0; nt < EMB / 16; ++nt) {
    v8f acc = wmma_tile_f32(Afeat, EMB, se_wr + nt * 16, EMB, EMB, lane);
    store_tile_lds(SeRow, EMB, acc, se_br, nt * 16, lane, true);
  }
  __syncthreads();

  for (int nt = 0; nt < EMB / 16; ++nt) {
    v8f acc = wmma_tile_f32(SeRow, EMB, se_we + nt * 16, EMB, EMB, lane);
    const int half = lane >> 4, l = lane & 15;
    const float bv = se_be[nt * 16 + l];
#pragma unroll
    for (int vv = 0; vv < 8; ++vv) {
      const int r = wv * 16 + vv + half * 8;
      const int col = nt * 16 + l;
      const float g = 1.0f / (1.0f + expf(-(acc[vv] + bv)));
      vposg[(size_t)(rowBase + r) * EMB + col] = sEmb[r * EMB + col] * g;
    }
  }
}

// ------------- 3x3 neighborhood attention: one wave per query --------------
__global__ void __launch_bounds__(256)
attn_kernel(const float* __restrict__ ref_pts, const float* __restrict__ qpos,
            const float* __restrict__ vposg, const float* __restrict__ ctr_feat,
            const int* __restrict__ vmap, float* __restrict__ out, int Q) {
  const int lane = threadIdx.x & 31;
  const int q = blockIdx.x * 8 + (threadIdx.x >> 5);
  if (q >= Q) return;

  const int bq  = (int)ref_pts[q * 3 + 0];
  const float x = ref_pts[q * 3 + 1];
  const float y = ref_pts[q * 3 + 2];
  const int rix = (int)(floorf(x / 0.2f) / 2.0f + 128.0f);  // metric2indices
  const int riy = (int)(floorf(y / 0.2f) / 2.0f + 128.0f);

  const float4 qv = *(const float4*)(qpos + (size_t)q * EMB + lane * 4);

  float s[9];
  int   iv[9];
#pragma unroll
  for (int n = 0; n < 9; ++n) {
    const int nx = rix + PADW + (n / 3 - 1);
    const int ny = riy + PADW + (n % 3 - 1);
    int idx = 0;
    if (bq >= 0 && bq < NB && nx >= 0 && nx < GP && ny >= 0 && ny < GP)
      idx = vmap[((size_t)bq * GP + nx) * GP + ny];
    iv[n] = idx;
    float p = 0.0f;
    if (idx > 0) {  // wave-uniform branch
      const float4 vv = *(const float4*)(vposg + (size_t)(idx - 1) * EMB + lane * 4);
      p = qv.x * vv.x + qv.y * vv.y + qv.z * vv.z + qv.w * vv.w;
    }
#pragma unroll
    for (int off = 16; off > 0; off >>= 1) p += __shfl_xor(p, off, 32);
    s[n] = p / 11.313708498984761f;  // missing -> 0, matches zero-row pad
  }

  float m = s[0];
#pragma unroll
  for (int n = 1; n < 9; ++n) m = fmaxf(m, s[n]);
  float w[9];
  float z = 0.0f;
#pragma unroll
  for (int n = 0; n < 9; ++n) { w[n] = expf(s[n] - m); z += w[n]; }
  const float rz = 1.0f / z;

  float ax = 0.f, ay = 0.f, az = 0.f, aw = 0.f;
#pragma unroll
  for (int n = 0; n < 9; ++n) {
    if (iv[n] > 0) {
      const float wn = w[n] * rz;
      const float4 f = *(const float4*)(ctr_feat + (size_t)(iv[n] - 1) * EMB + lane * 4);
      ax += wn * f.x; ay += wn * f.y; az += wn * f.z; aw += wn * f.w;
    }
  }
  *(float4*)(out + (size_t)q * EMB + lane * 4) = make_float4(ax, ay, az, aw);
}

extern "C" void kernel_launch(void* const* d_in, const int* in_sizes, int n_in,
                              void* d_out, int out_size, void* d_ws, size_t ws_size,
                              hipStream_t stream) {
  (void)n_in; (void)out_size; (void)ws_size;
  const float* ref_pts  = (const float*)d_in[0];
  const int*   ctr_coor = (const int*)d_in[1];
  const float* ctr_feat = (const float*)d_in[2];
  const float* q_w1 = (const float*)d_in[3];
  const float* q_b1 = (const float*)d_in[4];
  const float* q_w2 = (const float*)d_in[5];
  const float* q_b2 = (const float*)d_in[6];
  const float* v_w1 = (const float*)d_in[7];
  const float* v_b1 = (const float*)d_in[8];
  const float* v_w2 = (const float*)d_in[9];
  const float* v_b2 = (const float*)d_in[10];
  const float* se_wr = (const float*)d_in[11];
  const float* se_br = (const float*)d_in[12];
  const float* se_we = (const float*)d_in[13];
  const float* se_be = (const float*)d_in[14];
  float* out = (float*)d_out;

  const int Q = in_sizes[0] / 3;   // 65536
  const int V = in_sizes[1] / 3;   // 131072

  float* qpos  = (float*)d_ws;
  float* vposg = qpos + (size_t)Q * EMB;
  int*   vmap  = (int*)(vposg + (size_t)V * EMB);

  hipMemsetAsync(vmap, 0, sizeof(int) * (size_t)NB * GP * GP, stream);
  map_kernel<<<(V + 255) / 256, 256, 0, stream>>>(ctr_coor, vmap, V);
  qpos_kernel<<<Q / 32, 64, 0, stream>>>(ref_pts, q_w1, q_b1, q_w2, q_b2, qpos);
  vpos_kernel<<<V / 32, 64, 0, stream>>>(ctr_coor, ctr_feat, v_w1, v_b1, v_w2, v_b2,
                                         se_wr, se_br, se_we, se_be, vposg);
  attn_kernel<<<(Q + 7) / 8, 256, 0, stream>>>(ref_pts, qpos, vposg, ctr_feat, vmap, out, Q);
}